// MemorySlots_1254130451203
// MI455X (gfx1250) — compile-verified
//
#include <hip/hip_runtime.h>
#include <hip/hip_bf16.h>
#include <math.h>

#define BB 256
#define NN 2048
#define DD 128
#define TPB 256
#define NWAVES (TPB / 32)

typedef __attribute__((ext_vector_type(2))) float v2f;
typedef __attribute__((ext_vector_type(4))) float v4f;
typedef __attribute__((ext_vector_type(8))) float v8f;

// Output layout (floats), concatenated in reference return order.
static constexpr size_t OFF_READ  = 0;                                   // (B,D)
static constexpr size_t OFF_ATTN  = OFF_READ  + (size_t)BB * DD;         // (B,N)
static constexpr size_t OFF_MEM   = OFF_ATTN  + (size_t)BB * NN;         // (B,N,D)
static constexpr size_t OFF_AGE   = OFF_MEM   + (size_t)BB * NN * DD;    // (B,N)
static constexpr size_t OFF_USAGE = OFF_AGE   + (size_t)BB * NN;         // (B,N)
static constexpr size_t OFF_ALIVE = OFF_USAGE + (size_t)BB * NN;         // (B,N)
static constexpr size_t OFF_STATS = OFF_ALIVE + (size_t)BB * NN;         // 5 scalars

__global__ __launch_bounds__(TPB) void memslots_main(
    const float* __restrict__ query, const float* __restrict__ mem,
    const float* __restrict__ age_in, const float* __restrict__ usage_in,
    const int* __restrict__ alive_in, const int* __restrict__ retain_m,
    const int* __restrict__ update_m, const int* __restrict__ forget_m,
    const int* __restrict__ write_m, const float* __restrict__ update_vec,
    const float* __restrict__ write_vec, const int* __restrict__ ovw_idx,
    const float* __restrict__ emptyv, float* __restrict__ out,
    float* __restrict__ partials)
{
    __shared__ __align__(16) float s_q[DD];        // query row
    __shared__ __align__(16) float s_logits[NN];   // logits -> attn (8 KB)
    __shared__ float s_red[TPB];                   // scalar reductions
    __shared__ v4f   s_rv[TPB];                    // vec4 reductions

    const int t    = threadIdx.x;
    const int b    = blockIdx.x;
    const int lane = t & 31;
    const int wave = t >> 5;
    const size_t bN = (size_t)b * NN;

    if (t < DD) s_q[t] = query[b * DD + t];
    __syncthreads();

    // ---------------- Phase 1: logits via V_WMMA_F32_16X16X4_F32 -------------
    // A (16x4 fp32): lanes 0-15 = rows M, VGPR0/1 = K0/K1; lanes 16-31 = K2/K3.
    // B (4x16): query slice broadcast to all 16 columns, same K split as A.
    const int rowSel = lane & 15;
    const int kHalf  = (lane >> 4) * 2;  // 0 or 2
    for (int chunk = wave; chunk < NN / 16; chunk += NWAVES) {
        const int n0 = chunk * 16;
        const float* Ap = mem + (bN + n0 + rowSel) * DD + kHalf;
        const float* Bp = s_q + kHalf;
        v8f c0 = {}; v8f c1 = {};
        #pragma unroll
        for (int k = 0; k < DD; k += 8) {
            v2f a0 = *(const v2f*)(Ap + k);
            v2f a1 = *(const v2f*)(Ap + k + 4);
            v2f b0 = *(const v2f*)(Bp + k);
            v2f b1 = *(const v2f*)(Bp + k + 4);
            c0 = __builtin_amdgcn_wmma_f32_16x16x4_f32(false, a0, false, b0,
                                                       (short)0, c0, false, false);
            c1 = __builtin_amdgcn_wmma_f32_16x16x4_f32(false, a1, false, b1,
                                                       (short)0, c1, false, false);
        }
        v8f c = c0 + c1;
        // Column N=0 lives in lane 0 (M=0..7 in c[0..7]) and lane 16 (M=8..15).
        if (rowSel == 0) {
            const int m0 = (lane >> 4) * 8;
            #pragma unroll
            for (int r = 0; r < 8; ++r) {
                const int n = n0 + m0 + r;
                s_logits[n] = alive_in[bN + n] ? c[r] : -INFINITY;
            }
        }
    }
    __syncthreads();

    // ---------------- Phase 2: softmax + per-slot lifecycle ------------------
    float lv[NN / TPB];
    float lmax = -INFINITY;
    #pragma unroll
    for (int i = 0; i < NN / TPB; ++i) {
        lv[i] = s_logits[t + i * TPB];
        lmax = fmaxf(lmax, lv[i]);
    }
    s_red[t] = lmax; __syncthreads();
    for (int s = TPB / 2; s > 0; s >>= 1) {
        if (t < s) s_red[t] = fmaxf(s_red[t], s_red[t + s]);
        __syncthreads();
    }
    const float rowmax = s_red[0];
    __syncthreads();

    float lsum = 0.f;
    #pragma unroll
    for (int i = 0; i < NN / TPB; ++i) {
        const float e = (rowmax == -INFINITY) ? 0.f : expf(lv[i] - rowmax);
        lv[i] = e; lsum += e;
    }
    s_red[t] = lsum; __syncthreads();
    for (int s = TPB / 2; s > 0; s >>= 1) {
        if (t < s) s_red[t] += s_red[t + s];
        __syncthreads();
    }
    const float ssum = s_red[0];
    const float inv  = ssum > 0.f ? 1.f / ssum : 0.f;  // all-dead row -> attn 0

    const int tgt = min(max(ovw_idx[b], 0), NN - 1);
    const int wm  = write_m[b] != 0;

    float cA = 0.f, cG = 0.f, cU = 0.f, cF = 0.f;
    #pragma unroll
    for (int i = 0; i < NN / TPB; ++i) {
        const int n = t + i * TPB;
        const size_t idx = bN + n;
        const float at = lv[i] * inv;
        s_logits[n] = at;                               // keep attn for phase 3
        float us = 0.95f * usage_in[idx] + 0.05f * at;  // EMA before lifecycle
        int   alv = alive_in[idx] != 0;
        float ag  = age_in[idx] + (alv ? 1.0f : 0.0f);
        const int fg = forget_m[idx] != 0;
        const int up = update_m[idx] != 0;
        const int rt = retain_m[idx] != 0;
        if (fg) { alv = 0; ag = 0.f; us = 0.f; }
        if (up) { alv = 1; ag = 0.f; }
        if (rt) { alv = 1; }
        if (n == tgt && wm) { alv = 1; ag = 0.f; us = 0.f; }
        out[OFF_ATTN  + idx] = at;
        out[OFF_AGE   + idx] = ag;
        out[OFF_USAGE + idx] = us;
        out[OFF_ALIVE + idx] = alv ? 1.f : 0.f;
        cA += (float)alv;
        cG += alv ? ag : 0.f;
        cU += (float)up;
        cF += (float)fg;
    }
    {
        v4f p = {cA, cG, cU, cF};
        s_rv[t] = p;
    }
    __syncthreads();
    for (int s = TPB / 2; s > 0; s >>= 1) {
        if (t < s) s_rv[t] += s_rv[t + s];
        __syncthreads();
    }
    if (t == 0) {
        v4f p = s_rv[0];
        partials[b * 4 + 0] = p.x;
        partials[b * 4 + 1] = p.y;
        partials[b * 4 + 2] = p.z;
        partials[b * 4 + 3] = p.w;
    }
    __syncthreads();

    // ---------------- Phase 3: read_vec + mem rewrite (fused stream) ---------
    const int dgrp = t & 31;   // float4 column within D
    const int rgrp = t >> 5;   // 8 rows in flight
    const int dOff = dgrp * 4;
    const v4f uv = *(const v4f*)(update_vec + b * DD + dOff);
    const v4f wv = *(const v4f*)(write_vec  + b * DD + dOff);
    const v4f ev = *(const v4f*)(emptyv + dOff);
    v4f acc = {};
    for (int r = rgrp; r < NN; r += NWAVES) {
        const size_t idx = bN + r;
        const v4f m = *(const v4f*)(mem + idx * DD + dOff);
        acc += s_logits[r] * m;                       // read_vec uses OLD mem
        const int   fg = forget_m[idx] != 0;
        const float u  = (update_m[idx] != 0) ? 1.f : 0.f;
        v4f base = fg ? ev : m;
        base = (1.f - 0.5f * u) * base + (0.5f * u) * uv;
        if (r == tgt && wm) base = wv;
        __builtin_nontemporal_store(base, (v4f*)(out + OFF_MEM + idx * DD + dOff));
    }
    __syncthreads();
    s_rv[t] = acc;
    __syncthreads();
    if (rgrp < 4) s_rv[t] += s_rv[t + 4 * 32];
    __syncthreads();
    if (rgrp < 2) s_rv[t] += s_rv[t + 2 * 32];
    __syncthreads();
    if (rgrp < 1) s_rv[t] += s_rv[t + 32];
    __syncthreads();
    if (rgrp == 0) *(v4f*)(out + OFF_READ + b * DD + dOff) = s_rv[t];
}

__global__ __launch_bounds__(BB) void memslots_stats(
    const float* __restrict__ partials, const int* __restrict__ write_m,
    float* __restrict__ out)
{
    __shared__ v4f   s[BB];
    __shared__ float sw[BB];
    const int t = threadIdx.x;
    v4f p = {partials[t * 4 + 0], partials[t * 4 + 1],
             partials[t * 4 + 2], partials[t * 4 + 3]};
    s[t]  = p;
    sw[t] = (write_m[t] != 0) ? 1.f : 0.f;
    __syncthreads();
    for (int st = BB / 2; st > 0; st >>= 1) {
        if (t < st) { s[t] += s[t + st]; sw[t] += sw[t + st]; }
        __syncthreads();
    }
    if (t == 0) {
        v4f q = s[0];
        const float bn = (float)BB * (float)NN;
        out[OFF_STATS + 0] = q.x / bn;                  // utilization
        out[OFF_STATS + 1] = q.y / (q.x + 1e-6f);       // avg_age
        out[OFF_STATS + 2] = sw[0] / (float)BB;         // writes
        out[OFF_STATS + 3] = q.z / bn;                  // updates
        out[OFF_STATS + 4] = q.w / bn;                  // forgets
    }
}

extern "C" void kernel_launch(void* const* d_in, const int* in_sizes, int n_in,
                              void* d_out, int out_size, void* d_ws, size_t ws_size,
                              hipStream_t stream) {
    const float* query  = (const float*)d_in[0];
    const float* mem    = (const float*)d_in[1];
    const float* age    = (const float*)d_in[2];
    const float* usage  = (const float*)d_in[3];
    const int*   alive  = (const int*)d_in[4];
    const int*   retain = (const int*)d_in[5];
    const int*   update = (const int*)d_in[6];
    const int*   forget = (const int*)d_in[7];
    const int*   writem = (const int*)d_in[8];
    const float* uvec   = (const float*)d_in[9];
    const float* wvec   = (const float*)d_in[10];
    const int*   ovw    = (const int*)d_in[11];
    const float* emptyv = (const float*)d_in[12];
    float* out = (float*)d_out;
    float* partials = (float*)d_ws;   // 256*4 floats = 4 KB

    memslots_main<<<BB, TPB, 0, stream>>>(query, mem, age, usage, alive, retain,
                                          update, forget, writem, uvec, wvec,
                                          ovw, emptyv, out, partials);
    memslots_stats<<<1, BB, 0, stream>>>(partials, writem, out);
}